// GNNEncoder_12257836663105
// MI455X (gfx1250) — compile-verified
//
#include <hip/hip_runtime.h>

typedef __attribute__((ext_vector_type(2))) float v2f;
typedef __attribute__((ext_vector_type(8))) float v8f;

#define DIN 64
#define DH  128

// ---------------- zero workspace ----------------
__global__ void zero_kernel(float* __restrict__ p, long long n) {
    long long i = (long long)blockIdx.x * blockDim.x + threadIdx.x;
    long long stride = (long long)gridDim.x * blockDim.x;
    for (; i < n; i += stride) p[i] = 0.0f;
}

// ---------------- degree count ----------------
__global__ void count_kernel(const int* __restrict__ dst, float* __restrict__ cnt, int nedges) {
    int e = blockIdx.x * blockDim.x + threadIdx.x;
    if (e < nedges) atomicAdd(&cnt[dst[e]], 1.0f);
}

__global__ void inv_kernel(float* __restrict__ cnt, int n) {
    int i = blockIdx.x * blockDim.x + threadIdx.x;
    if (i < n) cnt[i] = 1.0f / fmaxf(cnt[i], 1.0f);
}

// ---------------- edge scatter-add (mean numerator) ----------------
// D/4 lanes per edge, float4 gather, 4 scalar f32 atomics (L2-resident target).
template<int D>
__global__ void scatter_kernel(const float* __restrict__ feat,
                               const int* __restrict__ src,
                               const int* __restrict__ dst,
                               float* __restrict__ agg,
                               int nedges) {
    constexpr int TPE = D / 4;  // 16 or 32
    long long tid = (long long)blockIdx.x * blockDim.x + threadIdx.x;
    long long e = tid / TPE;
    if (e >= nedges) return;
    int f = (int)(tid % TPE) * 4;
    int s = src[e];
    int d = dst[e];
    const float4 v = *(const float4*)(feat + (size_t)s * D + f);
    float* a = agg + (size_t)d * D + f;
    atomicAdd(a + 0, v.x);
    atomicAdd(a + 1, v.y);
    atomicAdd(a + 2, v.z);
    atomicAdd(a + 3, v.w);
}

// ---------------- fused SAGE GEMM: out = ReLU(inv*agg @ Wl^T + feat @ Wr^T + b) ----------------
// One wave computes one 16(M)x16(N) fp32 tile with V_WMMA_F32_16X16X4_F32.
// Block = 256 threads = 8 waves -> 16 rows x 128 cols per block. grid = N/16.
//
// Fragment layouts (ISA 7.12.2, wave32):
//   A 16x4 f32 : lane m=lane&15 is row; half=lane>>4 selects K {0,1} vs {2,3} -> float2
//   B 4x16 f32 : lane n=lane&15 is col; half selects rows {0,1} vs {2,3}      -> float2
//   C/D 16x16  : 8 VGPRs; elem r -> row r + 8*half, col = lane&15
template<int K>
__global__ __launch_bounds__(256) void sage_gemm_kernel(
    const float* __restrict__ agg,   // [N,K] mean numerator
    const float* __restrict__ feat,  // [N,K] self features
    const float* __restrict__ inv,   // [N]   1/max(deg,1)
    const float* __restrict__ Wl,    // [128,K] row-major
    const float* __restrict__ Wr,    // [128,K]
    const float* __restrict__ bias,  // [128]
    float* __restrict__ out,         // [N,128]
    int nrows) {
    const int lane = threadIdx.x & 31;
    const int wave = threadIdx.x >> 5;       // 0..7
    const int i0   = blockIdx.x * 16;        // M tile origin
    const int n0   = wave * 16;              // N tile origin

    const int m    = lane & 15;              // A row / B col / D col
    const int half = lane >> 4;              // 0 or 1
    const int krel = half * 2;               // intra-WMMA K offset

    int row = i0 + m;
    if (row >= nrows) row = nrows - 1;       // clamp: EXEC must stay all-ones for WMMA
    const float s = inv[row];

    const float* __restrict__ arow  = agg  + (size_t)row * K;
    const float* __restrict__ frow  = feat + (size_t)row * K;
    const float* __restrict__ wlrow = Wl + (size_t)(n0 + m) * K;
    const float* __restrict__ wrrow = Wr + (size_t)(n0 + m) * K;

    v8f acc = {};

    // mean-aggregated branch: (inv[i] * agg[i,:]) @ Wl^T
#pragma unroll
    for (int kk = 0; kk < K; kk += 4) {
        v2f a = *(const v2f*)(arow + kk + krel);
        a = a * s;
        v2f b = *(const v2f*)(wlrow + kk + krel);
        acc = __builtin_amdgcn_wmma_f32_16x16x4_f32(
            /*neg_a=*/false, a, /*neg_b=*/false, b,
            /*c_mod=*/(short)0, acc, /*reuse_a=*/false, /*reuse_b=*/false);
    }
    // self branch: feat[i,:] @ Wr^T
#pragma unroll
    for (int kk = 0; kk < K; kk += 4) {
        v2f a = *(const v2f*)(frow + kk + krel);
        v2f b = *(const v2f*)(wrrow + kk + krel);
        acc = __builtin_amdgcn_wmma_f32_16x16x4_f32(
            false, a, false, b, (short)0, acc, false, false);
    }

    const float bb = bias[n0 + m];
#pragma unroll
    for (int r = 0; r < 8; ++r) {
        int orow = i0 + r + half * 8;
        if (orow < nrows) {
            float v = acc[r] + bb;
            v = v > 0.0f ? v : 0.0f;  // ReLU
            out[(size_t)orow * DH + n0 + m] = v;
        }
    }
}

extern "C" void kernel_launch(void* const* d_in, const int* in_sizes, int n_in,
                              void* d_out, int out_size, void* d_ws, size_t ws_size,
                              hipStream_t stream) {
    const float* x   = (const float*)d_in[0];   // [N,64]
    const int*   ei  = (const int*)d_in[1];     // [2,E]
    const float* Wl1 = (const float*)d_in[2];   // [128,64]
    const float* Wr1 = (const float*)d_in[3];   // [128,64]
    const float* b1  = (const float*)d_in[4];   // [128]
    const float* Wl2 = (const float*)d_in[5];   // [128,128]
    const float* Wr2 = (const float*)d_in[6];   // [128,128]
    const float* b2  = (const float*)d_in[7];   // [128]
    float* out = (float*)d_out;                 // [N,128]

    const int N = in_sizes[0] / DIN;
    const int E = in_sizes[1] / 2;
    const int* srcv = ei;
    const int* dstv = ei + E;

    float* ws  = (float*)d_ws;
    float* inv = ws;                            // N floats (counts -> reciprocals)
    float* agg = ws + N;                        // N*128 floats (reused both layers)
    float* h1  = agg + (size_t)N * DH;          // N*128 floats

    const int ZB = 256;

    // --- degree counts (shared by both layers) + zero layer-1 accumulator ---
    {
        long long nz = (long long)N * (1 + DIN);  // inv + agg[N,64]
        int blocks = (int)((nz + ZB - 1) / ZB);
        zero_kernel<<<blocks, ZB, 0, stream>>>(ws, nz);
    }
    count_kernel<<<(E + ZB - 1) / ZB, ZB, 0, stream>>>(dstv, inv, E);
    inv_kernel<<<(N + ZB - 1) / ZB, ZB, 0, stream>>>(inv, N);

    // --- layer 1: scatter mean numerator, fused GEMM+bias+ReLU ---
    {
        long long nt = (long long)E * (DIN / 4);
        scatter_kernel<DIN><<<(int)((nt + ZB - 1) / ZB), ZB, 0, stream>>>(x, srcv, dstv, agg, E);
    }
    sage_gemm_kernel<DIN><<<(N + 15) / 16, 256, 0, stream>>>(agg, x, inv, Wl1, Wr1, b1, h1, N);

    // --- layer 2 ---
    {
        long long nz = (long long)N * DH;
        zero_kernel<<<(int)((nz + ZB - 1) / ZB), ZB, 0, stream>>>(agg, nz);
    }
    {
        long long nt = (long long)E * (DH / 4);
        scatter_kernel<DH><<<(int)((nt + ZB - 1) / ZB), ZB, 0, stream>>>(h1, srcv, dstv, agg, E);
    }
    sage_gemm_kernel<DH><<<(N + 15) / 16, 256, 0, stream>>>(agg, h1, inv, Wl2, Wr2, b2, out, N);
}